// MSAPeptideEmbedder_89902255440339
// MI455X (gfx1250) — compile-verified
//
#include <hip/hip_runtime.h>

// MSAPeptideEmbedder for MI455X (gfx1250, wave32).
// Output: [B=64, N=16, L=64, 512] fp32 = 128 MiB -> bandwidth bound (~5.5us @ 23.3 TB/s).
// aa_emb gather (channels 0..255) done as exact one-hot GEMM with V_WMMA_F32_16X16X4_F32.
// aa_emb staged into LDS by the Tensor Data Mover (tensor_load_to_lds, 6-arg form) with
// hardware row padding (256 dwords data + 8 dwords pad == AA_STRIDE 264), waited via
// s_wait_tensorcnt.

typedef float v2f __attribute__((ext_vector_type(2)));
typedef float v8f __attribute__((ext_vector_type(8)));
typedef unsigned int u32x4 __attribute__((ext_vector_type(4)));
typedef int i32x4 __attribute__((ext_vector_type(4)));
typedef int i32x8 __attribute__((ext_vector_type(8)));

#define B_ 64
#define N_ 16
#define L_ 64
#define DIM_ 512
#define VOCAB_ 28
#define AA_STRIDE 264   // 256 + 8 pad: avoids LDS bank overlap between lane halves

#define ENC_BASE 1.5915494309189535e-4f   // MIN_WL / (2*pi) = 0.001/(2pi)
#define LN_SCALE 16.118095650958320f      // ln(MAX_WL/MIN_WL) = ln(1e7)

#if defined(__has_builtin)
#  if __has_builtin(__builtin_amdgcn_tensor_load_to_lds) && \
      __has_builtin(__builtin_amdgcn_s_wait_tensorcnt)
#    define USE_TDM 1
#  endif
#endif
#ifndef USE_TDM
#  define USE_TDM 0
#endif

__global__ __launch_bounds__(512)
void MSAPeptideEmbedder_kernel(const int* __restrict__ tokens,
                               const float* __restrict__ precursors,
                               const float* __restrict__ mass_table,
                               const float* __restrict__ charge_emb,
                               const float* __restrict__ aa_emb,
                               float* __restrict__ out) {
    __shared__ float aa_lds[32 * AA_STRIDE];  // padded one-hot GEMM B operand
    __shared__ int   tok_lds[L_];
    __shared__ float pre_mass_lds[L_];
    __shared__ float suffix_lds[L_];
    __shared__ float prec_lds[256];

    const int tid = threadIdx.x;
    const int bn  = blockIdx.x;           // 0..1023  (b*N + n)
    const int b   = bn >> 4;              // bn / N_

    const float pmass   = precursors[b * 2 + 0];
    const float pcharge = precursors[b * 2 + 1];

    __builtin_prefetch(aa_emb + tid, 0, 0);   // global_prefetch_b8

    // ---- Stage 0: fill LDS ----------------------------------------------
    if (tid < L_) tok_lds[tid] = tokens[bn * L_ + tid];

#if USE_TDM
    // TDM: DMA aa_emb [28 x 256] f32 into LDS with +8-dword padding per
    // 256-dword row (hardware reproduces AA_STRIDE = 264). Wave 0 issues;
    // other waves overlap with prec/token work until the barrier.
    if (tid < 32) {
        const unsigned long long ga = (unsigned long long)(uintptr_t)aa_emb;
        const unsigned lds_off = (unsigned)(uintptr_t)(&aa_lds[0]); // low 32 bits = LDS byte offset
        u32x4 g0 = {
            1u,                                                   // count=1 (valid), user mode
            lds_off,                                              // [63:32]  lds_addr
            (unsigned)ga,                                         // [95:64]  global_addr lo
            ((unsigned)(ga >> 32) & 0x01FFFFFFu) | 0x80000000u    // [120:96] addr hi, [127:126] type=2
        };
        i32x8 g1 = {
            (int)((2u << 16) | (1u << 20) | (7u << 22) | (7u << 25)),
            //    data_size=4B | pad_enable | pad_interval=256dw | pad_amount=8dw
            (int)(256u << 16),   // [47:32] atomic_bar=0, [63:48] tensor_dim0.lo = 256
            (int)(28u  << 16),   // [79:64] tensor_dim0.hi=0, [95:80] tensor_dim1.lo = 28
            (int)(256u << 16),   // [111:96] tensor_dim1.hi=0, [127:112] tile_dim0 = 256
            28,                  // [143:128] tile_dim1 = 28, [159:144] tile_dim2 = 0
            256,                 // [191:160] tensor_dim0_stride.lo = 256
            0,                   // [207:192] stride.hi, [223:208] tensor_dim1_stride.lo = 0
            0
        };
        i32x4 gz4 = {0, 0, 0, 0};               // groups 2/3 unused (2D tensor)
        i32x8 gz8 = {0, 0, 0, 0, 0, 0, 0, 0};   // extra group (6-arg form), unused
        __builtin_amdgcn_tensor_load_to_lds(g0, g1, gz4, gz4, gz8, 0);
        __builtin_amdgcn_s_wait_tensorcnt(0);   // data in LDS before barrier signal
    }
#else
    // Fallback: manual coalesced copy of aa_emb rows 0..27, cols 0..255
    for (int idx = tid; idx < VOCAB_ * 256; idx += 512) {
        int r = idx >> 8, c = idx & 255;
        aa_lds[r * AA_STRIDE + c] = aa_emb[idx];
    }
#endif

    // zero-pad vocab rows 28..31 (one-hot A is 0 there; B must be finite 0.0)
    for (int idx = tid; idx < 4 * 256; idx += 512) {
        int r = 28 + (idx >> 8), c = idx & 255;
        aa_lds[r * AA_STRIDE + c] = 0.0f;
    }

    // prec row (256 entries): encode256(pmass) + charge_emb[charge-1]
    if (tid >= 64 && tid < 320) {
        int t = tid - 64;
        int i = t & 127;                                   // n_sin = n_cos = 128
        float term = ENC_BASE * expf((float)i * (LN_SCALE / 127.0f));
        float arg  = pmass / term;
        float e    = (t < 128) ? sinf(arg) : cosf(arg);
        int ci = (int)pcharge - 1;
        prec_lds[t] = e + charge_emb[ci * 256 + t];
    }
    __syncthreads();

    // ---- Stage 1: per-token masses --------------------------------------
    if (tid < L_) pre_mass_lds[tid] = mass_table[tok_lds[tid]];
    __syncthreads();

    // ---- Stage 2: suffix masses (sequential prefix per thread = cumsum order)
    if (tid < L_) {
        float s = 0.0f;
        for (int k = 0; k <= tid; ++k) s += pre_mass_lds[k];
        suffix_lds[tid] = pmass - s;
    }
    __syncthreads();

    // ---- Stage 3: channels 0..255 via one-hot x aa_emb, exact f32 WMMA --
    // 64 tiles of 16(l) x 16(d); 16 waves x 4 tiles each; K = 32 (padded vocab)
    {
        const int w    = tid >> 5;        // wave id 0..15
        const int lane = tid & 31;
        const int m    = lane & 15;       // row-in-tile / col-in-tile
        const int hs   = lane >> 4;       // lane half select
        const int lt   = w >> 2;          // l-tile 0..3
        const int dt0  = (w & 3) * 4;     // first d-tile of this wave

        const int l_row = lt * 16 + m;
        const int tokL  = tok_lds[l_row];

        const int l_base = lt * 16 + (hs ? 8 : 0);   // C-matrix M base for this half

        for (int j = 0; j < 4; ++j) {
            const int d_base = (dt0 + j) * 16;
            v8f c = {0.0f, 0.0f, 0.0f, 0.0f, 0.0f, 0.0f, 0.0f, 0.0f};
            #pragma unroll
            for (int kb = 0; kb < 32; kb += 4) {
                const int kA = kb + (hs ? 2 : 0);
                v2f a, bb;
                a.x = (tokL == kA)     ? 1.0f : 0.0f;   // one-hot A (16x4)
                a.y = (tokL == kA + 1) ? 1.0f : 0.0f;
                bb.x = aa_lds[kA       * AA_STRIDE + d_base + m];   // B (4x16)
                bb.y = aa_lds[(kA + 1) * AA_STRIDE + d_base + m];
                c = __builtin_amdgcn_wmma_f32_16x16x4_f32(
                        /*neg_a=*/false, a, /*neg_b=*/false, bb,
                        /*c_mod=*/(short)0, c, /*reuse_a=*/false, /*reuse_b=*/false);
            }
            const int d_out = d_base + m;
            const float padd = prec_lds[d_out];
            #pragma unroll
            for (int r = 0; r < 8; ++r) {
                const int l_out = l_base + r;
                float v = c[r] + ((l_out == 0) ? padd : 0.0f);
                out[((long)bn * L_ + l_out) * DIM_ + d_out] = v;
            }
        }
    }

    // ---- Stage 4: channels 256..511 (sinusoidal encodings) --------------
    {
        const int dd = tid & 255;           // channel - 256
        const int lg = tid >> 8;            // 0 -> rows 0..31, 1 -> rows 32..63
        const int i  = dd & 63;             // frequency index (n_sin = n_cos = 64)
        const float term = ENC_BASE * expf((float)i * (LN_SCALE / 63.0f));
        const bool iscos     = (dd & 64) != 0;
        const bool useSuffix = dd >= 128;

        const int l_start = lg * 32;
        for (int l = l_start; l < l_start + 32; ++l) {
            float x   = useSuffix ? suffix_lds[l] : pre_mass_lds[l];
            float arg = x / term;
            float v   = iscos ? cosf(arg) : sinf(arg);
            if (l == 0 && iscos) v += 1.0f;   // zero-encoding: sin(0)=0, cos(0)=1
            out[((long)bn * L_ + l) * DIM_ + 256 + dd] = v;
        }
    }
}

extern "C" void kernel_launch(void* const* d_in, const int* in_sizes, int n_in,
                              void* d_out, int out_size, void* d_ws, size_t ws_size,
                              hipStream_t stream) {
    const int*   tokens     = (const int*)d_in[0];
    const float* precursors = (const float*)d_in[1];
    const float* mass_table = (const float*)d_in[2];
    const float* charge_emb = (const float*)d_in[3];
    const float* aa_emb     = (const float*)d_in[4];
    float* out = (float*)d_out;

    dim3 grid(B_ * N_);   // 1024 workgroups, one per (b, n)
    dim3 block(512);      // 16 wave32 waves
    MSAPeptideEmbedder_kernel<<<grid, block, 0, stream>>>(
        tokens, precursors, mass_table, charge_emb, aa_emb, out);
}